// MultiHeadLatentAttention_28802050687485
// MI455X (gfx1250) — compile-verified
//
#include <hip/hip_runtime.h>
#include <hip/hip_bf16.h>

typedef __attribute__((ext_vector_type(2))) float v2f;
typedef __attribute__((ext_vector_type(8))) float v8f;

#define TOKENS 4096   // B*S
#define SEQ    2048
#define NHEADS 16
#define HDIM   128
#define HID    2048   // NHEADS*HDIM
#define KVRANK 512
#define ROPED  64
#define KVW    4096   // NHEADS*HDIM*2 (k|v interleaved per head)

__device__ __forceinline__ v8f wmma4(v2f a, v2f b, v8f c) {
  // D = A(16x4 f32) * B(4x16 f32) + C(16x16 f32)
  return __builtin_amdgcn_wmma_f32_16x16x4_f32(false, a, false, b, (short)0, c,
                                               false, false);
}

// Async memory->LDS copy, 16B per lane. VDST VGPR holds the LDS byte offset
// (= low 32 bits of the generic pointer to a __shared__ object). ASYNCcnt
// tracked; drain with s_wait_asynccnt.
__device__ __forceinline__ void cp_async_b128(void* lds, const void* gaddr) {
  asm volatile("global_load_async_to_lds_b128 %0, %1, off"
               :
               : "v"((unsigned)(unsigned long long)lds), "v"(gaddr)
               : "memory");
}
__device__ __forceinline__ void wait_async_le8() {
  asm volatile("s_wait_asynccnt 0x8" ::: "memory");
}
__device__ __forceinline__ void wait_async_0() {
  asm volatile("s_wait_asynccnt 0x0" ::: "memory");
}

// ---------------------------------------------------------------------------
// C[M,N] = A[M,K] @ W[N,K]^T (row-major). 64x64 tile per block, 4 waves; wave
// w computes rows w*16..w*16+15 across 4 N-subtiles. K staged 32-wide through
// double-buffered LDS via async loads (8 per thread per tile).
// ---------------------------------------------------------------------------
__global__ __launch_bounds__(128) void gemm_xwt_kernel(
    const float* __restrict__ A, const float* __restrict__ W,
    float* __restrict__ C, int M, int N, int K) {
  __shared__ float As[2][64][36];   // pad to 36 (16B-aligned, conflict-free)
  __shared__ float Ws[2][64][36];
  const int tid = threadIdx.x;
  const int wave = tid >> 5, lane = tid & 31;
  const int hf = lane >> 4, l16 = lane & 15;
  const int m0 = blockIdx.x * 64, n0 = blockIdx.y * 64;

  auto issue_tile = [&](int buf, int k0) {
#pragma unroll
    for (int j = 0; j < 4; ++j) {
      int idx = tid + j * 128;        // 512 float4 = 64 rows x 8 float4
      int r = idx >> 3;
      int c4 = (idx & 7) << 2;
      cp_async_b128(&As[buf][r][c4], &A[(size_t)(m0 + r) * K + k0 + c4]);
      cp_async_b128(&Ws[buf][r][c4], &W[(size_t)(n0 + r) * K + k0 + c4]);
    }
  };

  v8f acc[4];
#pragma unroll
  for (int nt = 0; nt < 4; ++nt)
#pragma unroll
    for (int v = 0; v < 8; ++v) acc[nt][v] = 0.0f;

  const int ntiles = K / 32;
  issue_tile(0, 0);
  for (int t = 0; t < ntiles; ++t) {
    const int buf = t & 1;
    if (t + 1 < ntiles) {
      issue_tile(buf ^ 1, (t + 1) * 32);
      wait_async_le8();   // drain previous 8 (in-order), keep next 8 in flight
    } else {
      wait_async_0();
    }
    __syncthreads();
    const int mb = wave * 16;
#pragma unroll
    for (int kk = 0; kk < 32; kk += 4) {
      v2f a;
      a.x = As[buf][mb + l16][kk + 2 * hf];
      a.y = As[buf][mb + l16][kk + 2 * hf + 1];
#pragma unroll
      for (int nt = 0; nt < 4; ++nt) {
        v2f b;
        b.x = Ws[buf][nt * 16 + l16][kk + 2 * hf];
        b.y = Ws[buf][nt * 16 + l16][kk + 2 * hf + 1];
        acc[nt] = wmma4(a, b, acc[nt]);
      }
    }
    __syncthreads();
  }
  const int mb = wave * 16;
#pragma unroll
  for (int nt = 0; nt < 4; ++nt)
#pragma unroll
    for (int v = 0; v < 8; ++v)
      C[(size_t)(m0 + mb + v + 8 * hf) * N + n0 + nt * 16 + l16] = acc[nt][v];
}

// ---------------------------------------------------------------------------
// RMSNorm over 512-wide rows, in place.
// ---------------------------------------------------------------------------
__global__ __launch_bounds__(256) void rmsnorm_kernel(
    float* __restrict__ x, const float* __restrict__ w) {
  __shared__ float red[256];
  const int t = threadIdx.x;
  float* xp = x + (size_t)blockIdx.x * KVRANK;
  float a = xp[t], b = xp[t + 256];
  red[t] = a * a + b * b;
  __syncthreads();
  for (int s = 128; s > 0; s >>= 1) {
    if (t < s) red[t] += red[t + s];
    __syncthreads();
  }
  float inv = rsqrtf(red[0] * (1.0f / KVRANK) + 1e-6f);
  xp[t] = w[t] * a * inv;
  xp[t + 256] = w[t + 256] * b * inv;
}

// ---------------------------------------------------------------------------
// YaRN RoPE, in place on q (per-head first 64 dims) and k-half of kv.
// One block per token; thread = head*32 + freq-index j.
// ---------------------------------------------------------------------------
__global__ __launch_bounds__(512) void rope_kernel(float* q, float* kv) {
  const int row = blockIdx.x;
  const int pos = row & (SEQ - 1);
  const int h = threadIdx.x >> 5;
  const int j = threadIdx.x & 31;

  const double TWO_PI = 6.283185307179586;
  double pf = pow(500000.0, (double)j / 32.0);
  double extrap = 1.0 / pf;
  double interp = 1.0 / (16.0 * pf);          // scaling = 131072/8192 = 16
  double l2b = 2.0 * log(500000.0);
  double lowd = floor(64.0 * log(8192.0 / (32.0 * TWO_PI)) / l2b);
  double highd = ceil(64.0 * log(8192.0 / (1.0 * TWO_PI)) / l2b);
  if (lowd < 0.0) lowd = 0.0;
  if (highd > 63.0) highd = 63.0;
  double denom = highd - lowd;
  if (denom < 1.0) denom = 1.0;
  double sm = ((double)j - lowd) / denom;
  sm = sm < 0.0 ? 0.0 : (sm > 1.0 ? 1.0 : sm);
  float invf = (float)((1.0 - sm) * interp + sm * extrap);

  float f = (float)pos * invf;
  float ms = (float)(0.1 * log(16.0) + 1.0);
  float c, s;
  sincosf(f, &s, &c);
  c *= ms;
  s *= ms;

  {
    float* qp = q + (size_t)row * HID + h * HDIM;
    float x = qp[j], y = qp[j + 32];
    qp[j] = x * c - y * s;
    qp[j + 32] = y * c + x * s;
  }
  {
    float* kp = kv + (size_t)row * KVW + h * 256;
    float x = kp[j], y = kp[j + 32];
    kp[j] = x * c - y * s;
    kp[j + 32] = y * c + x * s;
  }
}

// ---------------------------------------------------------------------------
// Flash attention: block = (64 q rows) x (1 head) x (1 batch). 4 waves; wave
// owns 16 q rows. Online softmax; Q resident in LDS; K/V async-staged per
// 16-key tile; P re-laid C->A through padded LDS.
// ---------------------------------------------------------------------------
__global__ __launch_bounds__(128) void mla_attn_kernel(const float* q,
                                                       const float* kv,
                                                       float* out) {
  __shared__ float Qs[64][132];
  __shared__ float Ks[16][132];
  __shared__ float Vs[16][132];
  __shared__ float Ps[4][16][20];
  const int tid = threadIdx.x;
  const int wave = tid >> 5, lane = tid & 31;
  const int hf = lane >> 4, l16 = lane & 15;
  const int head = blockIdx.y, bz = blockIdx.z;
  const int q0 = blockIdx.x * 64;
  const float scale = 0.08838834764831845f;  // 128^-0.5

  const float* qp = q + (size_t)bz * SEQ * HID;
  const float* kvp = kv + (size_t)bz * SEQ * KVW;

#pragma unroll
  for (int j = 0; j < 16; ++j) {
    int idx = tid + j * 128;   // 2048 float4 = 64 rows x 32 float4
    int r = idx >> 5;
    int c4 = (idx & 31) << 2;
    cp_async_b128(&Qs[r][c4],
                  &qp[(size_t)(q0 + r) * HID + head * HDIM + c4]);
  }

  v8f o[8];
  float m_i[8], l_i[8];
#pragma unroll
  for (int v = 0; v < 8; ++v) {
    m_i[v] = -1e30f;
    l_i[v] = 0.0f;
#pragma unroll
    for (int nt = 0; nt < 8; ++nt) o[nt][v] = 0.0f;
  }

  const int mb = wave * 16;
  for (int kt = 0; kt < SEQ / 16; ++kt) {
    __syncthreads();   // everyone done reading previous K/V tile
#pragma unroll
    for (int j = 0; j < 4; ++j) {
      int idx = tid + j * 128;   // 512 float4 = 16 rows x 32 float4
      int r = idx >> 5;
      int c4 = (idx & 31) << 2;
      const float* kb = &kvp[(size_t)(kt * 16 + r) * KVW + head * 256];
      cp_async_b128(&Ks[r][c4], &kb[c4]);
      cp_async_b128(&Vs[r][c4], &kb[128 + c4]);
    }
    wait_async_0();    // own copies (and, on kt==0, the Q tile) landed
    __syncthreads();   // all waves' copies landed

    // S = Q_tile(16x128) @ K_tile(16x128)^T
    v8f s;
#pragma unroll
    for (int v = 0; v < 8; ++v) s[v] = 0.0f;
#pragma unroll
    for (int kc = 0; kc < 128; kc += 4) {
      v2f a;
      a.x = Qs[mb + l16][kc + 2 * hf];
      a.y = Qs[mb + l16][kc + 2 * hf + 1];
      v2f b;
      b.x = Ks[l16][kc + 2 * hf];
      b.y = Ks[l16][kc + 2 * hf + 1];
      s = wmma4(a, b, s);
    }

    // online softmax: row r lives at VGPR v (r = v + 8*hf) across 16 lanes
#pragma unroll
    for (int v = 0; v < 8; ++v) {
      float x = s[v] * scale;
      float mx = x;
      mx = fmaxf(mx, __shfl_xor(mx, 1, 32));
      mx = fmaxf(mx, __shfl_xor(mx, 2, 32));
      mx = fmaxf(mx, __shfl_xor(mx, 4, 32));
      mx = fmaxf(mx, __shfl_xor(mx, 8, 32));
      float mnew = fmaxf(m_i[v], mx);
      float p = __expf(x - mnew);
      float ps = p;
      ps += __shfl_xor(ps, 1, 32);
      ps += __shfl_xor(ps, 2, 32);
      ps += __shfl_xor(ps, 4, 32);
      ps += __shfl_xor(ps, 8, 32);
      float alpha = __expf(m_i[v] - mnew);
      l_i[v] = l_i[v] * alpha + ps;
      m_i[v] = mnew;
#pragma unroll
      for (int nt = 0; nt < 8; ++nt) o[nt][v] *= alpha;
      Ps[wave][v + 8 * hf][l16] = p;   // C-layout -> LDS
    }

    // O += P(16x16) @ V_tile(16x128); DS ops are in-order per wave.
#pragma unroll
    for (int kc = 0; kc < 16; kc += 4) {
      v2f a;
      a.x = Ps[wave][l16][kc + 2 * hf];
      a.y = Ps[wave][l16][kc + 2 * hf + 1];
#pragma unroll
      for (int nt = 0; nt < 8; ++nt) {
        v2f b;
        b.x = Vs[kc + 2 * hf][nt * 16 + l16];
        b.y = Vs[kc + 2 * hf + 1][nt * 16 + l16];
        o[nt] = wmma4(a, b, o[nt]);
      }
    }
  }

#pragma unroll
  for (int v = 0; v < 8; ++v) {
    float inv = 1.0f / l_i[v];
    int row = q0 + mb + v + 8 * hf;
    float* op = out + ((size_t)(bz * SEQ + row)) * HID + head * HDIM;
#pragma unroll
    for (int nt = 0; nt < 8; ++nt) op[nt * 16 + l16] = o[nt][v] * inv;
  }
}

// ---------------------------------------------------------------------------
extern "C" void kernel_launch(void* const* d_in, const int* in_sizes, int n_in,
                              void* d_out, int out_size, void* d_ws,
                              size_t ws_size, hipStream_t stream) {
  (void)in_sizes; (void)n_in; (void)out_size; (void)ws_size;
  const float* hidden    = (const float*)d_in[0];
  const float* q_w       = (const float*)d_in[1];
  const float* kv_a_w    = (const float*)d_in[2];
  const float* kv_b_w    = (const float*)d_in[3];
  const float* o_w       = (const float*)d_in[4];
  const float* kv_norm_w = (const float*)d_in[5];
  float* out = (float*)d_out;
  float* ws = (float*)d_ws;

  float* qbuf = ws;                                  // TOKENS*HID   (33.5 MB)
  float* kvbuf = qbuf + (size_t)TOKENS * HID;        // TOKENS*KVW   (67 MB)
  float* latbuf = kvbuf + (size_t)TOKENS * KVW;      // TOKENS*KVRANK (8.4 MB)

  dim3 blk(128);
  // q = hidden @ q_w^T
  gemm_xwt_kernel<<<dim3(TOKENS / 64, HID / 64), blk, 0, stream>>>(
      hidden, q_w, qbuf, TOKENS, HID, HID);
  // latent = hidden @ kv_a_w[ROPED:,:]^T  (rope rows of kv_a are unused)
  gemm_xwt_kernel<<<dim3(TOKENS / 64, KVRANK / 64), blk, 0, stream>>>(
      hidden, kv_a_w + (size_t)ROPED * HID, latbuf, TOKENS, KVRANK, HID);
  // rmsnorm(latent)
  rmsnorm_kernel<<<dim3(TOKENS), dim3(256), 0, stream>>>(latbuf, kv_norm_w);
  // kv = latent @ kv_b_w^T
  gemm_xwt_kernel<<<dim3(TOKENS / 64, KVW / 64), blk, 0, stream>>>(
      latbuf, kv_b_w, kvbuf, TOKENS, KVW, KVRANK);
  // YaRN RoPE on q and k
  rope_kernel<<<dim3(TOKENS), dim3(512), 0, stream>>>(qbuf, kvbuf);
  // flash attention; output aliases qbuf (each block rewrites only the
  // q-region it consumed into LDS at block start)
  mla_attn_kernel<<<dim3(SEQ / 64, NHEADS, 2), blk, 0, stream>>>(qbuf, kvbuf,
                                                                 qbuf);
  // out = attn_out @ o_w^T
  gemm_xwt_kernel<<<dim3(TOKENS / 64, HID / 64), blk, 0, stream>>>(
      qbuf, o_w, out, TOKENS, HID, HID);
}